// CorotationalBeam2D_12146167513821
// MI455X (gfx1250) — compile-verified
//
#include <hip/hip_runtime.h>
#include <stdint.h>

#define BLK 256

// ---------------------------------------------------------------------------
// Kernel 1: per-node streaming maps + zero-init of the atomic scatter target.
//   o_nodal[i] = 0                         (RT store: stays L2-resident for
//                                           kernel 2's atomics)
//   o_fext[i]  = F_ext[i] / {F_c,F_c,M_c}  (NT in, NT out: touch-once)
//   o_phys[i]  = pred_raw[i]*{u_c,u_c,th_c}(NT out; pred_raw read RT here is
//                                           fine -- it IS the hot gather set,
//                                           this primes L2 for kernel 2)
// ---------------------------------------------------------------------------
__global__ __launch_bounds__(BLK) void beam_node_kernel(
    const float* __restrict__ pred_raw,
    const float* __restrict__ F_ext,
    const float* __restrict__ s_u, const float* __restrict__ s_th,
    const float* __restrict__ s_F, const float* __restrict__ s_M,
    float* __restrict__ o_nodal, float* __restrict__ o_fext,
    float* __restrict__ o_phys, int n3)
{
  int i = blockIdx.x * BLK + threadIdx.x;
  if (i >= n3) return;
  int comp = i % 3;
  float u_c = s_u[0], th_c = s_th[0], F_c = s_F[0], M_c = s_M[0];
  float dscale = (comp == 2) ? th_c : u_c;
  float fdiv   = (comp == 2) ? M_c  : F_c;

  float pr = pred_raw[i];                          // RT: primes hot gather set
  float fe = __builtin_nontemporal_load(&F_ext[i]); // touch-once stream

  __builtin_nontemporal_store(pr * dscale, &o_phys[i]);
  __builtin_nontemporal_store(fe / fdiv,   &o_fext[i]);
  o_nodal[i] = 0.0f;  // RT: keep resident for kernel 2's atomics
}

// ---------------------------------------------------------------------------
// Kernel 2: one thread per beam element.
//  - connectivity staged to LDS via gfx1250 async DMA (ASYNCcnt path);
//    independent coalesced NT property loads overlap the async copy.
//  - random gathers of coords/pred_raw rows (L2-resident hot set, RT).
//  - scatter-add nodal forces with global_atomic_add_f32 (L2 atomic units).
//  - all 7 per-element output streams stored NT (write-once, never re-read).
// ---------------------------------------------------------------------------
__global__ __launch_bounds__(BLK) void beam_elem_kernel(
    const float* __restrict__ pred_raw,
    const float* __restrict__ coords,
    const int*   __restrict__ conn,
    const float* __restrict__ prop_E,
    const float* __restrict__ prop_A,
    const float* __restrict__ prop_I,
    const float* __restrict__ s_u, const float* __restrict__ s_th,
    const float* __restrict__ s_F, const float* __restrict__ s_M,
    float* __restrict__ o_nodal,
    float* __restrict__ o_N,  float* __restrict__ o_M1,
    float* __restrict__ o_M2, float* __restrict__ o_V,
    float* __restrict__ o_l0, float* __restrict__ o_c,
    float* __restrict__ o_s,  int nElems)
{
  __shared__ int lds_conn[2 * BLK];

  const int tid = threadIdx.x;
  const int e   = blockIdx.x * BLK + tid;
  // Clamp so every lane issues valid loads (EXEC stays all-1s for the async op);
  // stores/atomics are predicated below.
  const int ec  = (e < nElems) ? e : (nElems - 1);

  // --- async DMA: connectivity pair (8B/lane) global -> LDS -----------------
  {
    uint64_t gaddr = (uint64_t)(uintptr_t)(conn + 2 * (size_t)ec);
    // low 32 bits of a flat LDS pointer == wave-relative LDS byte offset
    uint32_t laddr = (uint32_t)(uintptr_t)(&lds_conn[2 * tid]);
    asm volatile("global_load_async_to_lds_b64 %0, %1, off"
                 :: "v"(laddr), "v"(gaddr) : "memory");
  }

  // Independent coalesced touch-once loads overlap the in-flight async copy.
  float E = __builtin_nontemporal_load(&prop_E[ec]);
  float A = __builtin_nontemporal_load(&prop_A[ec]);
  float I = __builtin_nontemporal_load(&prop_I[ec]);
  float u_c = s_u[0], th_c = s_th[0], F_c = s_F[0], M_c = s_M[0];

  asm volatile("s_wait_asynccnt 0" ::: "memory");

  const int nA = lds_conn[2 * tid + 0];
  const int nB = lds_conn[2 * tid + 1];

  // --- geometry (hot L2-resident gathers, default RT) -----------------------
  const float* cA = coords + 3 * (size_t)nA;
  const float* cB = coords + 3 * (size_t)nB;
  float dx0 = cB[0] - cA[0];
  float dz0 = cB[2] - cA[2];
  float l0  = sqrtf(dx0 * dx0 + dz0 * dz0);
  float c   = dx0 / l0;
  float s   = dz0 / l0;

  // --- non-dimensional stiffness coefficients ------------------------------
  float EA = E * A, EI = E * I;
  float l0_2 = l0 * l0;
  float l0_3 = l0_2 * l0;
  float k_ax   = EA * u_c  / (F_c * l0);
  float k_bend = EI * th_c / (M_c * l0);
  float k_sw   = EI * th_c / (F_c * l0_2);
  float k_tr   = EI * u_c  / (F_c * l0_3);
  float k_mw   = EI * u_c  / (M_c * l0_2);

  // --- gather nodal dofs ----------------------------------------------------
  const float* pA = pred_raw + 3 * (size_t)nA;
  const float* pB = pred_raw + 3 * (size_t)nB;
  float pa0 = pA[0], pa1 = pA[1], pa2 = pA[2];
  float pb0 = pB[0], pb1 = pB[1], pb2 = pB[2];

  float ua =  c * pa0 + s * pa1;
  float wa = -s * pa0 + c * pa1;
  float ta = -pa2;
  float ub =  c * pb0 + s * pb1;
  float wb = -s * pb0 + c * pb1;
  float tb = -pb2;

  // --- local element forces (expressions kept literal vs. reference) -------
  float f0 = k_ax * (ua - ub);
  float f3 = k_ax * (ub - ua);
  float f1 = 12.0f * k_tr * (wa - wb) + 6.0f * k_sw * (ta + tb);
  float f4 = 12.0f * k_tr * (wb - wa) - 6.0f * k_sw * (ta + tb);
  float f2 = 6.0f * k_mw * (wa - wb) + k_bend * (4.0f * ta + 2.0f * tb);
  float f5 = 6.0f * k_mw * (wa - wb) + k_bend * (2.0f * ta + 4.0f * tb);

  if (e < nElems) {
    float gA0 = c * f0 - s * f1;
    float gA1 = s * f0 + c * f1;
    float gB0 = c * f3 - s * f4;
    float gB1 = s * f3 + c * f4;

    // scatter-add into nodal residual (L2-resident, f32 atomic units)
    atomicAdd(&o_nodal[3 * (size_t)nA + 0], gA0);
    atomicAdd(&o_nodal[3 * (size_t)nA + 1], gA1);
    atomicAdd(&o_nodal[3 * (size_t)nA + 2], f2);
    atomicAdd(&o_nodal[3 * (size_t)nB + 0], gB0);
    atomicAdd(&o_nodal[3 * (size_t)nB + 1], gB1);
    atomicAdd(&o_nodal[3 * (size_t)nB + 2], f5);

    // coalesced write-once output streams: NT so they don't evict the
    // hot gather/atomic working set from L2
    __builtin_nontemporal_store(f3 * F_c, &o_N[e]);
    __builtin_nontemporal_store(f2 * M_c, &o_M1[e]);
    __builtin_nontemporal_store(f5 * M_c, &o_M2[e]);
    __builtin_nontemporal_store(f4 * F_c, &o_V[e]);
    __builtin_nontemporal_store(l0,       &o_l0[e]);
    __builtin_nontemporal_store(c,        &o_c[e]);
    __builtin_nontemporal_store(s,        &o_s[e]);
  }
}

// ---------------------------------------------------------------------------
// Launcher
// Inputs (setup_inputs order):
//  0 pred_raw (N,3) f32   1 coords (N,3) f32   2 connectivity (E,2) i32
//  3 prop_E (E)           4 prop_A (E)         5 prop_I22 (E)
//  6 F_ext (N,3)          7 u_c  8 theta_c  9 F_c  10 M_c   (scalars)
// d_out layout (reference return order, flat):
//  nodal(3N) | F_ext_nd(3N) | N_e(E) | M1_e(E) | M2_e(E) | V_e(E)
//  | phys_disp(3N) | l0(E) | c(E) | s(E)
// ---------------------------------------------------------------------------
extern "C" void kernel_launch(void* const* d_in, const int* in_sizes, int n_in,
                              void* d_out, int out_size, void* d_ws, size_t ws_size,
                              hipStream_t stream)
{
  const float* pred_raw = (const float*)d_in[0];
  const float* coords   = (const float*)d_in[1];
  const int*   conn     = (const int*)  d_in[2];
  const float* prop_E   = (const float*)d_in[3];
  const float* prop_A   = (const float*)d_in[4];
  const float* prop_I   = (const float*)d_in[5];
  const float* F_ext    = (const float*)d_in[6];
  const float* s_u      = (const float*)d_in[7];
  const float* s_th     = (const float*)d_in[8];
  const float* s_F      = (const float*)d_in[9];
  const float* s_M      = (const float*)d_in[10];

  const int n3     = in_sizes[0];  // N_NODES * 3
  const int nElems = in_sizes[3];  // N_ELEMS

  float* out = (float*)d_out;
  const size_t nn3 = (size_t)n3;
  float* o_nodal = out;
  float* o_fext  = out + nn3;
  float* o_N     = out + 2 * nn3;
  float* o_M1    = o_N  + nElems;
  float* o_M2    = o_M1 + nElems;
  float* o_V     = o_M2 + nElems;
  float* o_phys  = o_V  + nElems;
  float* o_l0    = o_phys + nn3;
  float* o_c     = o_l0 + nElems;
  float* o_s     = o_c  + nElems;

  int gn = (n3 + BLK - 1) / BLK;
  beam_node_kernel<<<gn, BLK, 0, stream>>>(pred_raw, F_ext, s_u, s_th, s_F, s_M,
                                           o_nodal, o_fext, o_phys, n3);

  int ge = (nElems + BLK - 1) / BLK;
  beam_elem_kernel<<<ge, BLK, 0, stream>>>(pred_raw, coords, conn,
                                           prop_E, prop_A, prop_I,
                                           s_u, s_th, s_F, s_M,
                                           o_nodal, o_N, o_M1, o_M2, o_V,
                                           o_l0, o_c, o_s, nElems);
}